// DinoNet_48859547959329
// MI455X (gfx1250) — compile-verified
//
#include <hip/hip_runtime.h>
#include <stdint.h>

typedef __attribute__((ext_vector_type(2))) float v2f;
typedef __attribute__((ext_vector_type(4))) float v4f;
typedef __attribute__((ext_vector_type(8))) float v8f;

#define CH     1024
#define HF     192
#define WF     192
#define NPIX   (HF * WF)          // 36864
#define THRESH 0.2f
#define NEGV   (-1e30f)

// ---- order-preserving float<->u32 bijection (IEEE total order on u32) ----
__device__ __forceinline__ unsigned int mono_from_f(float f) {
    unsigned int u = __float_as_uint(f);
    return (u & 0x80000000u) ? ~u : (u | 0x80000000u);
}
__device__ __forceinline__ float f_from_mono(unsigned int m) {
    unsigned int u = (m & 0x80000000u) ? (m & 0x7FFFFFFFu) : ~m;
    return __uint_as_float(u);
}

// ---- zero the candidate counter ----
__global__ void init_kernel(unsigned int* cnt) { *cnt = 0u; }

// ---- Kernel A: response = sqrt(sum_c feat^2) via V_WMMA_F32_16X16X4_F32 ----
// One wave handles 64 consecutive pixels as 4 interleaved 16-pixel tiles:
// tile t owns pixels pbase + 4n + t (n = lane&15). Each lane streams the
// channel dimension with b128 loads of 4 consecutive pixels; float4 element t
// feeds accumulator t. A = ones(16x4), B = fsq(4x16): with A all-ones,
// D[m,n] = sum_k B[k,n] regardless of K-row placement, and the ISA's
// "row striped across lanes" B/D layout pins column n = lane&15.
__global__ void resp_kernel(const float* __restrict__ feat,
                            float* __restrict__ resp) {
    int tid  = blockIdx.x * blockDim.x + threadIdx.x;
    int wave = tid >> 5;
    int lane = tid & 31;
    int pbase = wave * 64;
    if (pbase >= NPIX) return;           // wave-uniform: EXEC stays all-ones

    int n    = lane & 15;                // tile column for this lane
    int crow = (lane >> 4) * 2;          // lanes 0-15 -> ch {c,c+1}; 16-31 -> {c+2,c+3}
    const float* base = feat + (size_t)crow * NPIX + pbase + 4 * n;

    v8f a0 = {}, a1 = {}, a2 = {}, a3 = {};
    v2f ones; ones.x = 1.0f; ones.y = 1.0f;

    for (int cb = 0; cb < CH; cb += 4) {
        v4f q0 = *(const v4f*)(base);          // row c0,   pixels 4n..4n+3
        v4f q1 = *(const v4f*)(base + NPIX);   // row c0+1, pixels 4n..4n+3
        __builtin_prefetch(base + (size_t)64 * NPIX, 0, 0);  // 16 chunks ahead
        base += (size_t)4 * NPIX;

        v2f b;
        b.x = q0[0] * q0[0]; b.y = q1[0] * q1[0];
        a0 = __builtin_amdgcn_wmma_f32_16x16x4_f32(false, ones, false, b,
                                                   (short)0, a0, false, false);
        b.x = q0[1] * q0[1]; b.y = q1[1] * q1[1];
        a1 = __builtin_amdgcn_wmma_f32_16x16x4_f32(false, ones, false, b,
                                                   (short)0, a1, false, false);
        b.x = q0[2] * q0[2]; b.y = q1[2] * q1[2];
        a2 = __builtin_amdgcn_wmma_f32_16x16x4_f32(false, ones, false, b,
                                                   (short)0, a2, false, false);
        b.x = q0[3] * q0[3]; b.y = q1[3] * q1[3];
        a3 = __builtin_amdgcn_wmma_f32_16x16x4_f32(false, ones, false, b,
                                                   (short)0, a3, false, false);
    }

    if (lane < 16) {
        v4f o;
        o[0] = __builtin_sqrtf(a0[0]);
        o[1] = __builtin_sqrtf(a1[0]);
        o[2] = __builtin_sqrtf(a2[0]);
        o[3] = __builtin_sqrtf(a3[0]);
        *(v4f*)(resp + pbase + 4 * n) = o;     // contiguous b128 store
    }
}

// ---- Kernel B: 9x9 max-pool NMS + threshold, compact candidates ----
__global__ void nms_kernel(const float* __restrict__ resp,
                           const int* __restrict__ nmsr_p,
                           float* __restrict__ masked,
                           unsigned long long* __restrict__ cand,
                           unsigned int* __restrict__ cnt) {
    int p = blockIdx.x * blockDim.x + threadIdx.x;
    if (p >= NPIX) return;
    int r = *nmsr_p;
    int x = p % WF, y = p / WF;
    float v = resp[p];
    float m = -__builtin_inff();
    for (int dy = -r; dy <= r; ++dy) {
        int yy = y + dy;
        if (yy < 0 || yy >= HF) continue;
        for (int dx = -r; dx <= r; ++dx) {
            int xx = x + dx;
            if (xx < 0 || xx >= WF) continue;
            m = fmaxf(m, resp[yy * WF + xx]);
        }
    }
    bool keep = (v > THRESH) && (v == m);
    masked[p] = keep ? v : NEGV;
    if (keep) {
        unsigned int pos = atomicAdd(cnt, 1u);   // pos < NPIX always
        unsigned long long key =
            ((unsigned long long)mono_from_f(v) << 32) |
            (unsigned long long)(~(unsigned int)p);  // score desc, idx asc
        cand[pos] = key;
    }
}

// ---- Kernel C: exact rank of each candidate; scatter into output slot ----
// Keys are all distinct -> ranks are a permutation -> deterministic output
// regardless of the atomic compaction order.
__global__ void topk_kernel(const unsigned long long* __restrict__ cand,
                            const unsigned int* __restrict__ cnt,
                            const int* __restrict__ imgh_p,
                            const int* __restrict__ imgw_p,
                            float* __restrict__ out, int K) {
    unsigned int n = *cnt;
    unsigned int i = blockIdx.x * blockDim.x + threadIdx.x;
    if (i >= n) return;
    unsigned long long k = cand[i];
    unsigned int rank = 0;
    for (unsigned int j = 0; j < n; ++j) rank += (cand[j] > k) ? 1u : 0u;
    if (rank < (unsigned int)K) {
        unsigned int p = ~(unsigned int)(k & 0xFFFFFFFFull);
        float score = f_from_mono((unsigned int)(k >> 32));
        float sx = (float)(*imgw_p) / (float)WF;
        float sy = (float)(*imgh_p) / (float)HF;
        out[2 * rank]     = (float)(p % WF) * sx;   // x
        out[2 * rank + 1] = (float)(p / WF) * sy;   // y
        out[2 * K + rank] = score;
    }
}

// ---- Kernel D: pad remaining slots like lax.top_k would (NEG ties -> smallest idx) ----
__global__ void pad_kernel(const float* __restrict__ masked,
                           const unsigned int* __restrict__ cnt,
                           const int* __restrict__ imgh_p,
                           const int* __restrict__ imgw_p,
                           float* __restrict__ out, int K) {
    if (threadIdx.x != 0 || blockIdx.x != 0) return;
    unsigned int n = *cnt;
    if (n >= (unsigned int)K) return;
    float sx = (float)(*imgw_p) / (float)WF;
    float sy = (float)(*imgh_p) / (float)HF;
    int slot = (int)n;
    for (int p = 0; p < NPIX && slot < K; ++p) {
        if (masked[p] == NEGV) {
            out[2 * slot]     = (float)(p % WF) * sx;
            out[2 * slot + 1] = (float)(p / WF) * sy;
            out[2 * K + slot] = NEGV;
            ++slot;
        }
    }
}

extern "C" void kernel_launch(void* const* d_in, const int* in_sizes, int n_in,
                              void* d_out, int out_size, void* d_ws, size_t ws_size,
                              hipStream_t stream) {
    const float* feat = (const float*)d_in[0];   // (1024,192,192) f32
    const int*   nmsr = (const int*)d_in[1];     // nms_radius
    const int*   imgh = (const int*)d_in[3];     // img_h
    const int*   imgw = (const int*)d_in[4];     // img_w
    float* out = (float*)d_out;
    int K = out_size / 3;                        // 256: xy(K,2) + scores(K)

    // workspace layout (256B-aligned base, 8B-aligned u64 array)
    char* ws = (char*)d_ws;
    unsigned int*       cnt    = (unsigned int*)ws;
    float*              resp   = (float*)(ws + 256);
    float*              masked = (float*)(ws + 256 + (size_t)NPIX * 4);
    unsigned long long* cand   = (unsigned long long*)(ws + 256 + (size_t)NPIX * 8);

    init_kernel<<<1, 1, 0, stream>>>(cnt);

    // 576 waves x 64 pixels = 36864 pixels; 256 threads = 8 waves per block
    resp_kernel<<<(NPIX / 64 * 32 + 255) / 256, 256, 0, stream>>>(feat, resp);

    nms_kernel<<<(NPIX + 255) / 256, 256, 0, stream>>>(resp, nmsr, masked, cand, cnt);

    topk_kernel<<<(NPIX + 255) / 256, 256, 0, stream>>>(cand, cnt, imgh, imgw, out, K);

    pad_kernel<<<1, 1, 0, stream>>>(masked, cnt, imgh, imgw, out, K);
}